// GAT_13683765805694
// MI455X (gfx1250) — compile-verified
//
#include <hip/hip_runtime.h>
#include <math.h>

#define F_IN   128
#define HID    64
#define HEADS  2
#define NC     40
#define NCP    48          // NC padded to a multiple of 16 for guard-free WMMA
#define NEG_SLOPE 0.2f

typedef float v2f __attribute__((ext_vector_type(2)));
typedef float v8f __attribute__((ext_vector_type(8)));

// ---- order-preserving float <-> uint mapping (for atomic segment-max) ----
__device__ __forceinline__ unsigned f32_ord(float f) {
  unsigned u = __float_as_uint(f);
  return (u & 0x80000000u) ? ~u : (u | 0x80000000u);
}
__device__ __forceinline__ float ord_f32(unsigned u) {
  unsigned b = (u & 0x80000000u) ? (u ^ 0x80000000u) : ~u;
  return __uint_as_float(b);
}

__device__ __forceinline__ void edge_sd(const int* __restrict__ srcI,
                                        const int* __restrict__ dstI,
                                        int E, int e, int& s, int& d) {
  if (e < E) { s = srcI[e]; d = dstI[e]; }
  else       { s = e - E;   d = e - E;   }   // self-loops appended at the end
}

// =======================================================================
// Guard-free fp32 WMMA GEMM: C[M,Ncols] = A[M,K] * B[K,Ncols].
// Requires: M % 16 == 0 (100000 = 6250*16), K % 4 == 0, Ncols % 16 == 0
// (layer 2 uses zero-padded B/C with Ncols = 48). One wave per 16x16 tile,
// V_WMMA_F32_16X16X4_F32, exact fp32 semantics.
// A layout (16x4 f32): lanes 0-15 hold K=k0,k0+1; lanes 16-31 hold K=k0+2,k0+3.
// B layout (4x16 f32): VGPR0 = rows k0 / k0+2, VGPR1 = rows k0+1 / k0+3.
// C layout: VGPR r -> row r (lanes 0-15), row r+8 (lanes 16-31).
// =======================================================================
__global__ void wmma_gemm_f32(const float* __restrict__ A,
                              const float* __restrict__ B,
                              float* __restrict__ C,
                              int M, int K, int Ncols, int tilesN)
{
  int wave = (blockIdx.x * blockDim.x + threadIdx.x) >> 5;
  int lane = threadIdx.x & 31;
  int tm = wave / tilesN;
  int tn = wave - tm * tilesN;
  if (tm * 16 >= M) return;                   // wave-uniform exit (EXEC stays full)

  const int half = lane >> 4;                 // 0 = lanes 0-15, 1 = lanes 16-31
  const int l15  = lane & 15;
  const int rowA = tm * 16 + l15;
  const int colB = tn * 16 + l15;

  const float* Ap = A + (size_t)rowA * K + half * 2;
  const float* Bp = B + colB;

  v8f acc = {0.f, 0.f, 0.f, 0.f, 0.f, 0.f, 0.f, 0.f};
  for (int k0 = 0; k0 < K; k0 += 4) {
    const int ka = k0 + half * 2;
    v2f a = *(const v2f*)(Ap + k0);                  // packed float2, 8B aligned
    v2f b;
    b.x = Bp[(size_t)ka * Ncols];
    b.y = Bp[(size_t)(ka + 1) * Ncols];
    acc = __builtin_amdgcn_wmma_f32_16x16x4_f32(false, a, false, b,
                                                (short)0, acc, false, false);
  }
  const int cm = tm * 16 + half * 8;
  float* Cp = C + (size_t)cm * Ncols + colB;
#pragma unroll
  for (int r = 0; r < 8; ++r)
    Cp[(size_t)r * Ncols] = acc[r];
}

// ---- zero-pad W2 [128,40] -> W2p [128,48] ----
__global__ void pad_w2_k(const float* __restrict__ W2, float* __restrict__ W2p)
{
  int i = blockIdx.x * blockDim.x + threadIdx.x;
  if (i >= (HEADS * HID) * NCP) return;
  int k = i / NCP;
  int c = i - k * NCP;
  W2p[i] = (c < NC) ? W2[k * NC + c] : 0.f;
}

// ---- layer-1 attention logits: al_s/al_d [N,2] from h1 [N,128], one wave/node ----
__global__ void node_logits1_k(const float* __restrict__ h,
                               const float* __restrict__ a_s,
                               const float* __restrict__ a_d,
                               float* __restrict__ al_s, float* __restrict__ al_d, int N)
{
  int wave = (blockIdx.x * blockDim.x + threadIdx.x) >> 5;
  int lane = threadIdx.x & 31;
  if (wave >= N) return;
  const float* row = h + (size_t)wave * (HEADS * HID);
  float v0 = row[lane],      v1 = row[lane + 32];
  float v2 = row[lane + 64], v3 = row[lane + 96];
  float s0 = v0 * a_s[lane] + v1 * a_s[lane + 32];
  float d0 = v0 * a_d[lane] + v1 * a_d[lane + 32];
  float s1 = v2 * a_s[lane + 64] + v3 * a_s[lane + 96];
  float d1 = v2 * a_d[lane + 64] + v3 * a_d[lane + 96];
#pragma unroll
  for (int off = 16; off > 0; off >>= 1) {
    s0 += __shfl_xor(s0, off, 32);  d0 += __shfl_xor(d0, off, 32);
    s1 += __shfl_xor(s1, off, 32);  d1 += __shfl_xor(d1, off, 32);
  }
  if (lane == 0) {
    al_s[wave * 2] = s0;  al_s[wave * 2 + 1] = s1;
    al_d[wave * 2] = d0;  al_d[wave * 2 + 1] = d1;
  }
}

// ---- layer-2 attention logits from padded h2p [N,48] (first 40 cols valid) ----
__global__ void node_logits2_k(const float* __restrict__ h,
                               const float* __restrict__ a_s,
                               const float* __restrict__ a_d,
                               float* __restrict__ al_s, float* __restrict__ al_d, int N)
{
  int wave = (blockIdx.x * blockDim.x + threadIdx.x) >> 5;
  int lane = threadIdx.x & 31;
  if (wave >= N) return;
  const float* row = h + (size_t)wave * NCP;
  float s = 0.f, d = 0.f;
  { float v = row[lane]; s = v * a_s[lane]; d = v * a_d[lane]; }   // lanes 0-31 < 40
  if (lane + 32 < NC) {
    float v = row[lane + 32];
    s += v * a_s[lane + 32];
    d += v * a_d[lane + 32];
  }
#pragma unroll
  for (int off = 16; off > 0; off >>= 1) {
    s += __shfl_xor(s, off, 32);  d += __shfl_xor(d, off, 32);
  }
  if (lane == 0) { al_s[wave] = s; al_d[wave] = d; }
}

// ---- segment-max over incoming edges (ordered-uint atomicMax) ----
__global__ void edge_max_k(const int* __restrict__ srcI, const int* __restrict__ dstI,
                           int E, int N, int H,
                           const float* __restrict__ al_s, const float* __restrict__ al_d,
                           unsigned* __restrict__ emax)
{
  int e = blockIdx.x * blockDim.x + threadIdx.x;
  if (e >= E + N) return;
  int s, d; edge_sd(srcI, dstI, E, e, s, d);
  for (int h = 0; h < H; ++h) {
    float v = al_s[s * H + h] + al_d[d * H + h];
    v = v > 0.f ? v : NEG_SLOPE * v;
    atomicMax(&emax[d * H + h], f32_ord(v));
  }
}

// ---- segment-sum of exp(e - max) ----
__global__ void edge_sum_k(const int* __restrict__ srcI, const int* __restrict__ dstI,
                           int E, int N, int H,
                           const float* __restrict__ al_s, const float* __restrict__ al_d,
                           const unsigned* __restrict__ emax, float* __restrict__ denom)
{
  int e = blockIdx.x * blockDim.x + threadIdx.x;
  if (e >= E + N) return;
  int s, d; edge_sd(srcI, dstI, E, e, s, d);
  for (int h = 0; h < H; ++h) {
    float v = al_s[s * H + h] + al_d[d * H + h];
    v = v > 0.f ? v : NEG_SLOPE * v;
    float m = ord_f32(emax[d * H + h]);
    atomicAdd(&denom[d * H + h], __expf(v - m));
  }
}

// ---- layer-1 aggregation: agg[dst] += alpha_h * h1[src], 128 cols, wave/edge ----
__global__ void edge_agg1_k(const int* __restrict__ srcI, const int* __restrict__ dstI,
                            int E, int N,
                            const float* __restrict__ al_s, const float* __restrict__ al_d,
                            const unsigned* __restrict__ emax, const float* __restrict__ denom,
                            const float* __restrict__ hsrc, float* __restrict__ agg)
{
  int wave = (blockIdx.x * blockDim.x + threadIdx.x) >> 5;
  int lane = threadIdx.x & 31;
  int nw   = (gridDim.x * blockDim.x) >> 5;
  const int Etot = E + N;
  for (int e = wave; e < Etot; e += nw) {
    int s, d; edge_sd(srcI, dstI, E, e, s, d);
    // prefetch next iteration's source row (4 x 128B lines) into cache
    int e2 = e + nw;
    if (e2 < Etot && lane < 4) {
      int s2, d2; edge_sd(srcI, dstI, E, e2, s2, d2);
      (void)d2;
      __builtin_prefetch(hsrc + (size_t)s2 * 128 + lane * 32, 0, 3);
    }
    float alpha0 = 0.f, alpha1 = 0.f;
#pragma unroll
    for (int h = 0; h < 2; ++h) {
      float v = al_s[s * 2 + h] + al_d[d * 2 + h];
      v = v > 0.f ? v : NEG_SLOPE * v;
      float m = ord_f32(emax[d * 2 + h]);
      float a = __expf(v - m) / (denom[d * 2 + h] + 1e-16f);
      if (h == 0) alpha0 = a; else alpha1 = a;
    }
    const float* hs = hsrc + (size_t)s * 128;
    float*       ad = agg  + (size_t)d * 128;
#pragma unroll
    for (int i = 0; i < 4; ++i) {
      int c = lane + 32 * i;
      float a = (c < HID) ? alpha0 : alpha1;   // head = c / 64
      atomicAdd(&ad[c], a * hs[c]);
    }
  }
}

// ---- layer-2 aggregation: out[dst] += alpha * h2p[src], 40 cols, wave/edge ----
__global__ void edge_agg2_k(const int* __restrict__ srcI, const int* __restrict__ dstI,
                            int E, int N,
                            const float* __restrict__ al_s, const float* __restrict__ al_d,
                            const unsigned* __restrict__ emax, const float* __restrict__ denom,
                            const float* __restrict__ hsrc, float* __restrict__ agg)
{
  int wave = (blockIdx.x * blockDim.x + threadIdx.x) >> 5;
  int lane = threadIdx.x & 31;
  int nw   = (gridDim.x * blockDim.x) >> 5;
  const int Etot = E + N;
  for (int e = wave; e < Etot; e += nw) {
    int s, d; edge_sd(srcI, dstI, E, e, s, d);
    int e2 = e + nw;
    if (e2 < Etot && lane < 2) {
      int s2, d2; edge_sd(srcI, dstI, E, e2, s2, d2);
      (void)d2;
      __builtin_prefetch(hsrc + (size_t)s2 * NCP + lane * 32, 0, 3);
    }
    float v = al_s[s] + al_d[d];
    v = v > 0.f ? v : NEG_SLOPE * v;
    float m = ord_f32(emax[d]);
    float alpha = __expf(v - m) / (denom[d] + 1e-16f);
    const float* hs = hsrc + (size_t)s * NCP;   // padded stride
    float*       ad = agg  + (size_t)d * NC;    // d_out stride = 40
    atomicAdd(&ad[lane], alpha * hs[lane]);                    // lanes 0-31 < 40
    if (lane + 32 < NC) atomicAdd(&ad[lane + 32], alpha * hs[lane + 32]);
  }
}

// ---- layer-1 epilogue: h_act = ELU(agg + b1) (C = 128) ----
__global__ void bias_elu_k(const float* __restrict__ agg, const float* __restrict__ b,
                           float* __restrict__ out, int total)
{
  int i = blockIdx.x * blockDim.x + threadIdx.x;
  if (i >= total) return;
  float v = agg[i] + b[i & 127];
  out[i] = v > 0.f ? v : expm1f(v);
}

// ---- layer-2 epilogue: out = log_softmax(out + b2), one wave/node ----
__global__ void logsoftmax_k(float* __restrict__ out, const float* __restrict__ b2, int N)
{
  int wave = (blockIdx.x * blockDim.x + threadIdx.x) >> 5;
  int lane = threadIdx.x & 31;
  if (wave >= N) return;
  float* o = out + (size_t)wave * NC;
  float v0 = o[lane] + b2[lane];                                 // lanes 0-31 < 40
  float v1 = (lane + 32 < NC) ? o[lane + 32] + b2[lane + 32] : -3.4e38f;
  float mx = fmaxf(v0, v1);
#pragma unroll
  for (int off = 16; off > 0; off >>= 1) mx = fmaxf(mx, __shfl_xor(mx, off, 32));
  float se = __expf(v0 - mx) + ((lane + 32 < NC) ? __expf(v1 - mx) : 0.f);
#pragma unroll
  for (int off = 16; off > 0; off >>= 1) se += __shfl_xor(se, off, 32);
  float lse = mx + __logf(se);
  o[lane] = v0 - lse;
  if (lane + 32 < NC) o[lane + 32] = v1 - lse;
}

extern "C" void kernel_launch(void* const* d_in, const int* in_sizes, int n_in,
                              void* d_out, int out_size, void* d_ws, size_t ws_size,
                              hipStream_t stream)
{
  const float* x      = (const float*)d_in[0];
  const int*   eidx   = (const int*)  d_in[1];
  const float* W1     = (const float*)d_in[2];
  const float* a_src1 = (const float*)d_in[3];
  const float* a_dst1 = (const float*)d_in[4];
  const float* b1     = (const float*)d_in[5];
  const float* W2     = (const float*)d_in[6];
  const float* a_src2 = (const float*)d_in[7];
  const float* a_dst2 = (const float*)d_in[8];
  const float* b2     = (const float*)d_in[9];
  float* out = (float*)d_out;

  const int N = in_sizes[0] / F_IN;     // 100000
  const int E = in_sizes[1] / 2;        // 1600000
  const int Etot = E + N;               // + self-loops
  const int* srcI = eidx;
  const int* dstI = eidx + E;

  // ---- workspace carve (~126 MB of fp32) ----
  float*    h1     = (float*)d_ws;                        // N*128
  float*    agg1   = h1     + (size_t)N * 128;            // N*128
  float*    al_s1  = agg1   + (size_t)N * 128;            // N*2
  float*    al_d1  = al_s1  + (size_t)N * 2;              // N*2
  unsigned* emax1  = (unsigned*)(al_d1 + (size_t)N * 2);  // N*2
  float*    denom1 = (float*)(emax1 + (size_t)N * 2);     // N*2
  float*    h2p    = denom1 + (size_t)N * 2;              // N*48 (padded)
  float*    al_s2  = h2p    + (size_t)N * NCP;            // N
  float*    al_d2  = al_s2  + N;                          // N
  unsigned* emax2  = (unsigned*)(al_d2 + N);              // N
  float*    denom2 = (float*)(emax2 + N);                 // N
  float*    W2p    = denom2 + N;                          // 128*48

  hipMemsetAsync(agg1,   0, (size_t)N * 128 * sizeof(float),    stream);
  hipMemsetAsync(emax1,  0, (size_t)N * 2   * sizeof(unsigned), stream);
  hipMemsetAsync(denom1, 0, (size_t)N * 2   * sizeof(float),    stream);
  hipMemsetAsync(emax2,  0, (size_t)N       * sizeof(unsigned), stream);
  hipMemsetAsync(denom2, 0, (size_t)N       * sizeof(float),    stream);
  hipMemsetAsync(out,    0, (size_t)N * NC  * sizeof(float),    stream);

  const int BLK = 256;                        // 8 waves / block
  const int nodeWaveGrid = (N * 32 + BLK - 1) / BLK;
  const int edgeGrid     = (Etot + BLK - 1) / BLK;

  pad_w2_k<<<((HEADS * HID) * NCP + BLK - 1) / BLK, BLK, 0, stream>>>(W2, W2p);

  // ===== layer 1 =====
  {
    int waves = (N / 16) * ((HEADS * HID) / 16);   // 6250 * 8
    int grid  = (waves * 32 + BLK - 1) / BLK;
    wmma_gemm_f32<<<grid, BLK, 0, stream>>>(x, W1, h1, N, F_IN, HEADS * HID, (HEADS * HID) / 16);
  }
  node_logits1_k<<<nodeWaveGrid, BLK, 0, stream>>>(h1, a_src1, a_dst1, al_s1, al_d1, N);
  edge_max_k<<<edgeGrid, BLK, 0, stream>>>(srcI, dstI, E, N, HEADS, al_s1, al_d1, emax1);
  edge_sum_k<<<edgeGrid, BLK, 0, stream>>>(srcI, dstI, E, N, HEADS, al_s1, al_d1, emax1, denom1);
  edge_agg1_k<<<2048, BLK, 0, stream>>>(srcI, dstI, E, N, al_s1, al_d1, emax1, denom1, h1, agg1);
  bias_elu_k<<<(N * 128 + BLK - 1) / BLK, BLK, 0, stream>>>(agg1, b1, h1, N * 128);

  // ===== layer 2 (guard-free via zero-padded W2p / h2p, Ncols = 48) =====
  {
    int waves = (N / 16) * (NCP / 16);             // 6250 * 3
    int grid  = (waves * 32 + BLK - 1) / BLK;
    wmma_gemm_f32<<<grid, BLK, 0, stream>>>(h1, W2p, h2p, N, HEADS * HID, NCP, NCP / 16);
  }
  node_logits2_k<<<nodeWaveGrid, BLK, 0, stream>>>(h2p, a_src2, a_dst2, al_s2, al_d2, N);
  edge_max_k<<<edgeGrid, BLK, 0, stream>>>(srcI, dstI, E, N, 1, al_s2, al_d2, emax2);
  edge_sum_k<<<edgeGrid, BLK, 0, stream>>>(srcI, dstI, E, N, 1, al_s2, al_d2, emax2, denom2);
  edge_agg2_k<<<2048, BLK, 0, stream>>>(srcI, dstI, E, N, al_s2, al_d2, emax2, denom2, h2p, out);
  logsoftmax_k<<<nodeWaveGrid, BLK, 0, stream>>>(out, b2, N);
}